// GraphConvLayer_49357764165671
// MI455X (gfx1250) — compile-verified
//
#include <hip/hip_runtime.h>

// CDNA5 / gfx1250: wave32, WMMA (not MFMA).
typedef __attribute__((ext_vector_type(2))) float v2f;
typedef __attribute__((ext_vector_type(8))) float v8f;

#define F_DIM      128
#define LDS_STRIDE 132   // 132 % 64 == 4 -> 16 rows at same col hit 16 distinct banks

// -----------------------------------------------------------------------------
// Fused dual GEMM: self_out = feat @ w_s ; nbr_out = feat @ w_n
// Block = 256 threads = 8 wave32; block handles 16 rows; wave w handles cols
// [16w, 16w+16). K loop in steps of 4 using V_WMMA_F32_16X16X4_F32 (f32 exact).
// -----------------------------------------------------------------------------
__global__ __launch_bounds__(256) void gemm_dual_wmma(
    const float* __restrict__ feat,
    const float* __restrict__ w_s,
    const float* __restrict__ w_n,
    float* __restrict__ self_out,
    float* __restrict__ nbr_out,
    int n_rows)
{
    __shared__ float tileA[16 * LDS_STRIDE];

    const int tid  = threadIdx.x;
    const int base = blockIdx.x * 16;
    if (base >= n_rows) return;

    // Cooperative stage of the 16x128 feature strip (float4 per thread, x2).
    for (int i = tid; i < 16 * (F_DIM / 4); i += 256) {
        const int r  = i >> 5;        // 32 float4 per row
        const int c4 = i & 31;
        float4 v = make_float4(0.f, 0.f, 0.f, 0.f);
        if (base + r < n_rows)
            v = ((const float4*)(feat + (size_t)(base + r) * F_DIM))[c4];
        float* p = &tileA[r * LDS_STRIDE + c4 * 4];
        p[0] = v.x; p[1] = v.y; p[2] = v.z; p[3] = v.w;
    }
    __syncthreads();

    const int wave    = tid >> 5;
    const int lane    = tid & 31;
    const int colbase = wave * 16;
    const int m       = lane & 15;   // A row within tile / B,D column within tile
    const int kh      = lane >> 4;   // K-pair select (ISA 16x4 f32 A layout)

    v8f acc_s = {};
    v8f acc_n = {};

#pragma unroll 8
    for (int k = 0; k < F_DIM; k += 4) {
        // A fragment: lane holds A[m][k+2*kh], A[m][k+2*kh+1] (8B-aligned LDS pair)
        const v2f a = *(const v2f*)&tileA[m * LDS_STRIDE + k + 2 * kh];

        // B fragments: lane holds B[k+2*kh][col], B[k+2*kh+1][col]
        const int krow = k + 2 * kh;
        v2f bs, bn;
        bs.x = w_s[(size_t)(krow    ) * F_DIM + colbase + m];
        bs.y = w_s[(size_t)(krow + 1) * F_DIM + colbase + m];
        bn.x = w_n[(size_t)(krow    ) * F_DIM + colbase + m];
        bn.y = w_n[(size_t)(krow + 1) * F_DIM + colbase + m];

        acc_s = __builtin_amdgcn_wmma_f32_16x16x4_f32(
            false, a, false, bs, (short)0, acc_s, false, false);
        acc_n = __builtin_amdgcn_wmma_f32_16x16x4_f32(
            false, a, false, bn, (short)0, acc_n, false, false);
    }

    // C/D layout: VGPR r -> row (r + 8*kh), col (lane&15).
    // Block-uniform full-strip test -> scalar branch; hot path stores with no
    // per-row exec masking (all 6250 blocks take it when N % 16 == 0).
    if (base + 16 <= n_rows) {
        const size_t obase = (size_t)(base + 8 * kh) * F_DIM + colbase + m;
#pragma unroll
        for (int r = 0; r < 8; ++r) {
            self_out[obase + (size_t)r * F_DIM] = acc_s[r];
            nbr_out [obase + (size_t)r * F_DIM] = acc_n[r];
        }
    } else {
#pragma unroll
        for (int r = 0; r < 8; ++r) {
            const int row = base + r + 8 * kh;
            if (row < n_rows) {
                self_out[(size_t)row * F_DIM + colbase + m] = acc_s[r];
                nbr_out [(size_t)row * F_DIM + colbase + m] = acc_n[r];
            }
        }
    }
}

// -----------------------------------------------------------------------------
// Edge scatter: one wave32 per edge. Lane l moves float4 #l of the 128-float
// row: coalesced 512B global_load_b128 gather + 4 global_atomic_add_f32.
// nbr_conv (51MB) and out (51MB) are both L2-resident (192MB L2), so the
// 1.6GB of irregular traffic resolves at L2 atomic units, not HBM.
// -----------------------------------------------------------------------------
__global__ __launch_bounds__(256) void edge_scatter(
    const float* __restrict__ nbr,
    const int*   __restrict__ src,
    const int*   __restrict__ dst,
    float*       __restrict__ out,
    int n_edges)
{
    const int gwave = (int)((blockIdx.x * 256u + threadIdx.x) >> 5);
    const int lane  = threadIdx.x & 31;
    if (gwave >= n_edges) return;

    // Wave-uniform edge endpoints -> force SGPR so row bases use saddr form.
    const int s = __builtin_amdgcn_readfirstlane(src[gwave]);
    const int d = __builtin_amdgcn_readfirstlane(dst[gwave]);

    const float4 v = ((const float4*)(nbr + (size_t)s * F_DIM))[lane];
    float* o = out + (size_t)d * F_DIM + lane * 4;
    atomicAdd(o + 0, v.x);
    atomicAdd(o + 1, v.y);
    atomicAdd(o + 2, v.z);
    atomicAdd(o + 3, v.w);
}

// -----------------------------------------------------------------------------
// In-place ReLU, float4 vectorized.
// -----------------------------------------------------------------------------
__global__ __launch_bounds__(256) void relu_inplace(float* __restrict__ out, int n4)
{
    const int i = (int)(blockIdx.x * 256u + threadIdx.x);
    if (i >= n4) return;
    float4 v = ((float4*)out)[i];
    v.x = fmaxf(v.x, 0.f);
    v.y = fmaxf(v.y, 0.f);
    v.z = fmaxf(v.z, 0.f);
    v.w = fmaxf(v.w, 0.f);
    ((float4*)out)[i] = v;
}

// -----------------------------------------------------------------------------
// Inputs (setup_inputs order): features[N,128] f32, w_s[128,128] f32,
// w_n[128,128] f32, src[E] i32, dst[E] i32. Output: [N,128] f32.
// Workspace: nbr_conv [N,128] f32 (51.2 MB).
// -----------------------------------------------------------------------------
extern "C" void kernel_launch(void* const* d_in, const int* in_sizes, int n_in,
                              void* d_out, int out_size, void* d_ws, size_t ws_size,
                              hipStream_t stream)
{
    (void)n_in; (void)out_size; (void)ws_size;

    const float* feat = (const float*)d_in[0];
    const float* w_s  = (const float*)d_in[1];
    const float* w_n  = (const float*)d_in[2];
    const int*   src  = (const int*)d_in[3];
    const int*   dst  = (const int*)d_in[4];
    float*       out  = (float*)d_out;
    float*       nbr  = (float*)d_ws;

    const int n_rows  = in_sizes[0] / F_DIM;
    const int n_edges = in_sizes[3];

    // Phase 1: d_out = feat @ w_s ; ws = feat @ w_n  (features read once)
    const int strips = (n_rows + 15) / 16;
    gemm_dual_wmma<<<strips, 256, 0, stream>>>(feat, w_s, w_n, out, nbr, n_rows);

    // Phase 2: d_out[dst[e]] += ws[src[e]] for every edge (1 wave per edge)
    const int eblocks = (n_edges + 7) / 8;
    edge_scatter<<<eblocks, 256, 0, stream>>>(nbr, src, dst, out, n_edges);

    // Phase 3: d_out = relu(d_out)
    const int n4 = n_rows * (F_DIM / 4);
    relu_inplace<<<(n4 + 255) / 256, 256, 0, stream>>>(out, n4);
}